// TACEOptimizationField_90658169684158
// MI455X (gfx1250) — compile-verified
//
#include <hip/hip_runtime.h>
#include <stdint.h>

typedef float v8f __attribute__((ext_vector_type(8)));
typedef float v2f __attribute__((ext_vector_type(2)));

#define ALPHA_F   0.85f
#define EPS_F     1e-6f
#define SEG_CAP_F 50.0f
#define SKEL_IT   10
#define UNC_CLAMP 2.0f

// ---------------------------------------------------------------------------
// Wave-wide fp32 sum on the matrix pipe (v_wmma_f32_16x16x4_f32, B = ones):
// A holds the 32 per-lane partials; D row-sums + 7 VALU adds + one xor-16
// shuffle give the full 32-lane sum in every lane. EXEC must be all ones.
// ---------------------------------------------------------------------------
static __device__ __forceinline__ float wave_sum32(float p) {
    v2f a; a[0] = p;    a[1] = 0.0f;
    v2f b; b[0] = 1.0f; b[1] = 1.0f;
    v8f c = {0.f, 0.f, 0.f, 0.f, 0.f, 0.f, 0.f, 0.f};
    c = __builtin_amdgcn_wmma_f32_16x16x4_f32(
            /*neg_a=*/false, a, /*neg_b=*/false, b,
            /*c_mod=*/(short)0, c, /*reuse_a=*/false, /*reuse_b=*/false);
    float s = c[0] + c[1] + c[2] + c[3] + c[4] + c[5] + c[6] + c[7];
    s += __shfl_xor(s, 16, 32);
    return s;
}

static __device__ __forceinline__ float clamp01(float v) {
    return fminf(fmaxf(v, 0.0f), 1.0f);
}

// ---------------------------------------------------------------------------
// Zero the scalar accumulator block (deterministic per call).
// ---------------------------------------------------------------------------
__global__ void k_zero(float* __restrict__ scal, int n) {
    int i = blockIdx.x * blockDim.x + threadIdx.x;
    if (i < n) scal[i] = 0.0f;
}

// ---------------------------------------------------------------------------
// Pass 1: focal BCE partial sums (WMMA wave-reduced) + build fp16 state.
// ---------------------------------------------------------------------------
__global__ void k_focal_init(const float* __restrict__ logits,
                             const float* __restrict__ mgt,
                             const float* __restrict__ wsdf,
                             _Float16* __restrict__ prob,
                             _Float16* __restrict__ gt,
                             _Float16* __restrict__ curr_p,
                             _Float16* __restrict__ curr_g,
                             _Float16* __restrict__ skel_p,
                             _Float16* __restrict__ skel_g,
                             float* __restrict__ scal,
                             int n) {
    int tid    = blockIdx.x * blockDim.x + threadIdx.x;
    int stride = gridDim.x * blockDim.x;
    float accL = 0.0f, accW = 0.0f;
    for (int i = tid; i < n; i += stride) {
        float x = logits[i];
        float g = clamp01(mgt[i]);
        float w = clamp01(wsdf[i]);
        float p = 1.0f / (1.0f + __expf(-x));          // sigmoid
        float pc = fminf(fmaxf(p, EPS_F), 1.0f - EPS_F);
        float pt = pc * g + (1.0f - pc) * (1.0f - g);
        float om = 1.0f - pt;
        float focal = (ALPHA_F * g + (1.0f - ALPHA_F) * (1.0f - g)) * om * om;
        float bce = -(g * __logf(pc) + (1.0f - g) * __logf(1.0f - pc));
        accL += focal * bce * w;
        accW += w;
        _Float16 ph = (_Float16)p;
        _Float16 gh = (_Float16)g;
        prob[i]   = ph;
        gt[i]     = gh;
        curr_p[i] = ph;
        curr_g[i] = gh;
        skel_p[i] = (_Float16)0.0f;
        skel_g[i] = (_Float16)0.0f;
    }
    float sL = wave_sum32(accL);
    float sW = wave_sum32(accW);
    if ((threadIdx.x & 31) == 0) {
        atomicAdd(&scal[0], sL);
        atomicAdd(&scal[1], sW);
    }
}

// ---------------------------------------------------------------------------
// One fused skeleton iteration (erode + open + skel update):
//   ctile (curr, +2 halo) -> etile (eroded, +1 halo, -inf outside image)
//   opened = 3x3 max(etile);  delta = relu(curr - opened)
//   skel  += relu(delta - skel*delta);  erod_out = eroded center
// Interior blocks stage ctile via global_load_async_to_lds_b32 (ASYNCcnt);
// image-border blocks use clamped scalar loads. blockIdx.z = batch*2 + image.
// ---------------------------------------------------------------------------
#define OBX 32
#define OBY 8
__global__ void k_skel_iter(const _Float16* __restrict__ curr_p,
                            const _Float16* __restrict__ curr_g,
                            _Float16* __restrict__ erod_p,
                            _Float16* __restrict__ erod_g,
                            _Float16* __restrict__ skel_p,
                            _Float16* __restrict__ skel_g,
                            int H, int W) {
    __shared__ _Float16 ctile[OBY + 4][OBX + 4];   // 12 x 36 fp16
    __shared__ float    etile[OBY + 2][OBX + 2];   // 10 x 34 f32

    int slice = blockIdx.z;
    int img = slice & 1;
    size_t base = (size_t)(slice >> 1) * H * W;
    const _Float16* cu = (img ? curr_g : curr_p) + base;
    _Float16*       er = (img ? erod_g : erod_p) + base;
    _Float16*       sk = (img ? skel_g : skel_p) + base;

    int x0 = blockIdx.x * OBX;
    int y0 = blockIdx.y * OBY;
    int tid = threadIdx.y * OBX + threadIdx.x;

    bool interior = (x0 >= 2) && (x0 + OBX + 2 <= W) &&
                    (y0 >= 2) && (y0 + OBY + 2 <= H);
    if (interior) {
        // 12 rows x 18 dwords (36 fp16) staged async into LDS
        const int ROWW = (OBX + 4) / 2;            // 18 dwords per row
        const int NW   = (OBY + 4) * ROWW;         // 216 dwords
        if (tid < NW) {
            int row = tid / ROWW;
            int cw  = tid - row * ROWW;
            const _Float16* gp =
                cu + (size_t)(y0 - 2 + row) * W + (x0 - 2) + cw * 2;
            unsigned lds = (unsigned)(uintptr_t)&ctile[row][cw * 2];
            unsigned long long ga = (unsigned long long)(uintptr_t)gp;
            asm volatile("global_load_async_to_lds_b32 %0, %1, off"
                         :: "v"(lds), "v"(ga) : "memory");
        }
        asm volatile("s_wait_asynccnt 0x0" ::: "memory");
    } else {
        for (int t = tid; t < (OBY + 4) * (OBX + 4); t += OBX * OBY) {
            int cy = t / (OBX + 4);
            int cx = t - cy * (OBX + 4);
            int gy = min(max(y0 - 2 + cy, 0), H - 1);
            int gx = min(max(x0 - 2 + cx, 0), W - 1);
            ctile[cy][cx] = cu[(size_t)gy * W + gx];
        }
    }
    __syncthreads();

    // eroded = min(vert-3 min, horiz-3 min); -inf outside the image so the
    // dilate max reproduces reference SAME/-inf padding exactly.
    for (int t = tid; t < (OBY + 2) * (OBX + 2); t += OBX * OBY) {
        int ty = t / (OBX + 2);
        int tx = t - ty * (OBX + 2);
        int ey = y0 - 1 + ty, ex = x0 - 1 + tx;
        float e;
        if (ey < 0 || ey >= H || ex < 0 || ex >= W) {
            e = -1e30f;
        } else {
            int cy = ty + 1, cx = tx + 1;
            float c = (float)ctile[cy][cx];
            float u = (float)ctile[cy - 1][cx];
            float d = (float)ctile[cy + 1][cx];
            float l = (float)ctile[cy][cx - 1];
            float r = (float)ctile[cy][cx + 1];
            e = fminf(fminf(fminf(u, d), c), fminf(fminf(l, r), c));
        }
        etile[ty][tx] = e;
    }
    __syncthreads();

    int lx = threadIdx.x, ly = threadIdx.y;
    float m = etile[ly][lx];
    m = fmaxf(m, etile[ly][lx + 1]);
    m = fmaxf(m, etile[ly][lx + 2]);
    m = fmaxf(m, etile[ly + 1][lx]);
    m = fmaxf(m, etile[ly + 1][lx + 1]);
    m = fmaxf(m, etile[ly + 1][lx + 2]);
    m = fmaxf(m, etile[ly + 2][lx]);
    m = fmaxf(m, etile[ly + 2][lx + 1]);
    m = fmaxf(m, etile[ly + 2][lx + 2]);

    float ec = etile[ly + 1][lx + 1];               // eroded center
    float c  = (float)ctile[ly + 2][lx + 2];        // curr center
    size_t gi = (size_t)(y0 + ly) * W + (x0 + lx);
    float s = (float)sk[gi];
    float delta = fmaxf(c - m, 0.0f);
    s = s + fmaxf(delta - s * delta, 0.0f);
    sk[gi] = (_Float16)s;
    er[gi] = (_Float16)ec;                          // next iteration's curr
}

// ---------------------------------------------------------------------------
// Per-batch clDice sums (WMMA wave-reduced). blockIdx.y = batch index.
// ---------------------------------------------------------------------------
__global__ void k_reduce(const _Float16* __restrict__ prob,
                         const _Float16* __restrict__ gt,
                         const _Float16* __restrict__ skel_p,
                         const _Float16* __restrict__ skel_g,
                         float* __restrict__ scal,
                         int HW) {
    int b = blockIdx.y;
    size_t base = (size_t)b * HW;
    int tid    = blockIdx.x * blockDim.x + threadIdx.x;
    int stride = gridDim.x * blockDim.x;
    float a0 = 0.f, a1 = 0.f, a2 = 0.f, a3 = 0.f;
    for (int i = tid; i < HW; i += stride) {
        float sp = clamp01((float)skel_p[base + i]);
        float sg = clamp01((float)skel_g[base + i]);
        float g  = (float)gt[base + i];
        float p  = (float)prob[base + i];
        a0 += sp * g;
        a1 += sp;
        a2 += sg * p;
        a3 += sg;
    }
    a0 = wave_sum32(a0);
    a1 = wave_sum32(a1);
    a2 = wave_sum32(a2);
    a3 = wave_sum32(a3);
    if ((threadIdx.x & 31) == 0) {
        atomicAdd(&scal[2  + b], a0);
        atomicAdd(&scal[10 + b], a1);
        atomicAdd(&scal[18 + b], a2);
        atomicAdd(&scal[26 + b], a3);
    }
}

// ---------------------------------------------------------------------------
// Final scalar combine.
// ---------------------------------------------------------------------------
__global__ void k_final(const float* __restrict__ scal,
                        const float* __restrict__ log_vars,
                        float* __restrict__ out,
                        int B) {
    if (threadIdx.x == 0 && blockIdx.x == 0) {
        float lossSeg = scal[0] / fmaxf(scal[1], 1.0f);
        lossSeg = fminf(lossSeg, SEG_CAP_F);
        float acc = 0.0f;
        for (int b = 0; b < B; ++b) {
            float tp = scal[2  + b] / (scal[10 + b] + 1e-6f);
            float ts = scal[18 + b] / (scal[26 + b] + 1e-6f);
            acc += 2.0f * tp * ts / (tp + ts + 1e-6f);
        }
        float lossTopo = 1.0f - acc / (float)B;
        float s0 = fminf(fmaxf(log_vars[0], -UNC_CLAMP), UNC_CLAMP);
        float s1 = fminf(fmaxf(log_vars[1], -UNC_CLAMP), UNC_CLAMP);
        out[0] = __expf(-s0) * lossSeg + 0.5f * s0
               + __expf(-s1) * lossTopo + 0.5f * s1;
    }
}

// ---------------------------------------------------------------------------
// Workspace: 256 B scalar block + 8 fp16 arrays of N (prob, gt, curr_p,
// curr_g, erod_p, erod_g, skel_p, skel_g) = 128 MB -> L2-resident iteration.
// ---------------------------------------------------------------------------
extern "C" void kernel_launch(void* const* d_in, const int* in_sizes, int n_in,
                              void* d_out, int out_size, void* d_ws, size_t ws_size,
                              hipStream_t stream) {
    const float* logits  = (const float*)d_in[0];
    const float* mgt     = (const float*)d_in[1];
    const float* wsdf    = (const float*)d_in[2];
    const float* logvars = (const float*)d_in[3];
    float* out = (float*)d_out;

    const int B = 8, H = 1024, W = 1024;
    const int N = B * H * W;

    float* scal = (float*)d_ws;
    _Float16* arena = (_Float16*)((char*)d_ws + 256);
    _Float16* prob = arena + 0 * (size_t)N;
    _Float16* gt   = arena + 1 * (size_t)N;
    _Float16* cp   = arena + 2 * (size_t)N;
    _Float16* cg   = arena + 3 * (size_t)N;
    _Float16* ep   = arena + 4 * (size_t)N;
    _Float16* eg   = arena + 5 * (size_t)N;
    _Float16* sp   = arena + 6 * (size_t)N;
    _Float16* sg   = arena + 7 * (size_t)N;

    k_zero<<<1, 64, 0, stream>>>(scal, 34);

    // 1,048,576 threads -> exactly 8 px/thread (uniform loops for WMMA EXEC)
    k_focal_init<<<4096, 256, 0, stream>>>(logits, mgt, wsdf,
                                           prob, gt, cp, cg, sp, sg, scal, N);

    dim3 skB(OBX, OBY), skG(W / OBX, H / OBY, B * 2);
    for (int it = 0; it < SKEL_IT; ++it) {
        k_skel_iter<<<skG, skB, 0, stream>>>(cp, cg, ep, eg, sp, sg, H, W);
        _Float16* t;
        t = cp; cp = ep; ep = t;   // next curr = this iteration's eroded
        t = cg; cg = eg; eg = t;
    }

    // 131,072 threads/batch -> exactly 8 px/thread (uniform)
    k_reduce<<<dim3(512, B), 256, 0, stream>>>(prob, gt, sp, sg, scal, H * W);

    k_final<<<1, 32, 0, stream>>>(scal, logvars, out, B);
}